// BaseDecoder_36180804501646
// MI455X (gfx1250) — compile-verified
//
#include <hip/hip_runtime.h>
#include <stdint.h>

// BaseDecoder for MI455X (gfx1250).
//
// Roofline: 132 B read + 4 B write per element, 2.1M elements => ~285 MB at
// 23.3 TB/s => ~12 us floor. The decode must therefore cost only a few
// instructions per element. Key trick: v_cmp+ballot packs all 32 digit bits
// of one element in ONE instruction (lane l tests digit 31-l, ballot bit l
// lands at packed bit l). One v_writelane_b32 parks element j's word in lane
// j. ~3 wave-instructions/element vs ~115 for a per-lane shift-or decode.
//
// Data movement uses the CDNA5 async engine: global->LDS b128 copies
// (ASYNCcnt) staging a 256-element / 33 KB tile, then conflict-free LDS reads
// (digit row = contiguous reversed 128 B; sign column stride 33, gcd(33,64)=1).

#define ROW_DWORDS   33                          // 1 sign flag + 32 digits
#define THREADS      256
#define ELEMS_BLK    256
#define TILE_DWORDS  (ELEMS_BLK * ROW_DWORDS)    // 8448 dwords = 33792 B
#define B128_ROUNDS  8                           // 8 * 256 thr * 16 B = 32768 B
#define ROUND_BYTES  (THREADS * 16)              // 4096 B per b128 round
#define TAIL_OFFSET  (B128_ROUNDS * ROUND_BYTES) // 32768 B; tail = 1024 B via b32

__global__ __launch_bounds__(THREADS)
void base_decoder_kernel(const uint32_t* __restrict__ in,
                         uint32_t* __restrict__ out,
                         int num_elems)
{
    __shared__ __align__(16) uint32_t smem[TILE_DWORDS];

    const int t    = threadIdx.x;
    const int lane = t & 31;
    const int wave = t >> 5;
    const long long tile_base = (long long)blockIdx.x * ELEMS_BLK;

    if (tile_base + ELEMS_BLK <= num_elems) {
        // -------- async b128 global -> LDS staging (inst offset hits BOTH) ----
        const uint32_t g0  = (uint32_t)((uint64_t)tile_base * (ROW_DWORDS * 4));
        const uint32_t ls0 = (uint32_t)(uintptr_t)(&smem[0]);
        uint32_t g16 = g0 + t * 16, l16 = ls0 + t * 16;   // 16B-aligned lanes
        #pragma unroll
        for (int i = 0; i < B128_ROUNDS; ++i) {
            asm volatile("global_load_async_to_lds_b128 %0, %1, %2 offset:%3"
                         :
                         : "v"(l16), "v"(g16), "s"(in), "i"(i * ROUND_BYTES)
                         : "memory");
        }
        uint32_t g4 = g0 + t * 4, l4 = ls0 + t * 4;       // last 256 dwords
        asm volatile("global_load_async_to_lds_b32 %0, %1, %2 offset:%3"
                     :
                     : "v"(l4), "v"(g4), "s"(in), "i"(TAIL_OFFSET)
                     : "memory");
        asm volatile("s_wait_asynccnt 0" ::: "memory");
        __syncthreads();

        // -------- wave-ballot decode: wave w owns elements [w*32, w*32+32) ----
        const float* smem_f = (const float*)smem;
        const int ebase = wave * 32;
        // lane l tests digit (31-l) => dword (32-l) of each element's row
        const int dig0 = ebase * ROW_DWORDS + (32 - lane);
        uint32_t acc = 0u;
        #pragma unroll
        for (int j = 0; j < 32; ++j) {
            float f = smem_f[dig0 + j * ROW_DWORDS];            // ds_load_b32
            uint32_t b = __builtin_amdgcn_ballot_w32(f > 0.5f); // v_cmp -> sgpr
            // acc[lane j] = b  (b is wave-uniform, lives in an SGPR already)
            asm volatile("v_writelane_b32 %0, %1, %2"
                         : "+v"(acc)
                         : "s"(b), "i"(j));
        }
        // sign: lane handles element ebase+lane; ch0 in {0.0f,1.0f} -> bit 23
        uint32_t sbits = smem[(ebase + lane) * ROW_DWORDS];
        uint32_t res = acc ^ ((sbits & 0x00800000u) << 8);      // flip sign bit
        out[tile_base + ebase + lane] = res;                    // coalesced b32
    } else {
        // -------- tail path: direct per-element loads (exact, rarely taken) ----
        const long long e = tile_base + t;
        if (e < num_elems) {
            const uint32_t* q = in + (uint64_t)e * ROW_DWORDS;
            uint32_t neg    = (q[0] >> 23) & 1u;
            uint32_t packed = 0u;
            #pragma unroll
            for (int k = 1; k <= 32; ++k) {
                packed = (packed << 1) | ((q[k] >> 23) & 1u);
            }
            out[e] = packed ^ (neg << 31);
        }
    }
}

extern "C" void kernel_launch(void* const* d_in, const int* in_sizes, int n_in,
                              void* d_out, int out_size, void* d_ws, size_t ws_size,
                              hipStream_t stream)
{
    (void)n_in; (void)d_ws; (void)ws_size; (void)in_sizes;
    const uint32_t* in  = (const uint32_t*)d_in[0];   // raw float32 bits
    uint32_t*       out = (uint32_t*)d_out;           // bitcast into float32 out

    const int num_elems = out_size;                   // B*N
    const int blocks = (num_elems + ELEMS_BLK - 1) / ELEMS_BLK;

    hipLaunchKernelGGL(base_decoder_kernel, dim3(blocks), dim3(THREADS), 0, stream,
                       in, out, num_elems);
}